// SymbolicPlanner_24678882083293
// MI455X (gfx1250) — compile-verified
//
#include <hip/hip_runtime.h>

typedef float v2f __attribute__((ext_vector_type(2)));
typedef float v8f __attribute__((ext_vector_type(8)));

#define GRID_STRIDE 19

// ---------------------------------------------------------------------------
// Kernel P: repack weights into WMMA B-fragment order and zero-pad.
// Fragment layout for a [K][N] weight (K->mult of 4, N->mult of 16):
//   frag[a][t][s][lane][c] = W[a][4s + 2*(lane>>4) + c][16t + (lane&15)]
// so each lane's two K-values are adjacent -> one coalesced global_load_b64.
//   W1 [3][6][100]   -> W1f [3][7][2][32][2]    (pads k>=6, n>=100 with 0)
//   W2 [3][100][400] -> W2f [3][25][25][32][2]  (exact, no pad)
//   W3 [3][400][100] -> W3f [3][7][100][32][2]  (pads n>=100)
//   W4 [3][100][6]   -> W4f [3][1][25][32][2]   (pads n>=6)
//   b1,b3 -> [3][112] padded; b4 -> [3][16] padded; b2 used raw.
// ---------------------------------------------------------------------------
__global__ __launch_bounds__(256) void sp_pack_weights(
    const float* __restrict__ W1, const float* __restrict__ b1,
    const float* __restrict__ W2,
    const float* __restrict__ W3, const float* __restrict__ b3,
    const float* __restrict__ W4, const float* __restrict__ b4,
    float* __restrict__ W1f, float* __restrict__ b1p,
    float* __restrict__ W2f,
    float* __restrict__ W3f, float* __restrict__ b3p,
    float* __restrict__ W4f, float* __restrict__ b4p)
{
  const int tid = blockIdx.x * blockDim.x + threadIdx.x;

  if (tid < 3 * 7 * 100 * 64) {                    // W3f
    int c = tid & 1, l = (tid >> 1) & 31, rest = tid >> 6;
    int s = rest % 100; rest /= 100;
    int t = rest % 7, a = rest / 7;
    int k = 4 * s + 2 * (l >> 4) + c;              // < 400 always
    int n = 16 * t + (l & 15);
    W3f[tid] = (n < 100) ? W3[(a * 400 + k) * 100 + n] : 0.f;
  }
  if (tid < 3 * 25 * 25 * 64) {                    // W2f (exact dims)
    int c = tid & 1, l = (tid >> 1) & 31, rest = tid >> 6;
    int s = rest % 25; rest /= 25;
    int t = rest % 25, a = rest / 25;
    int k = 4 * s + 2 * (l >> 4) + c;
    int n = 16 * t + (l & 15);
    W2f[tid] = W2[(a * 100 + k) * 400 + n];
  }
  if (tid < 3 * 7 * 2 * 64) {                      // W1f
    int c = tid & 1, l = (tid >> 1) & 31, rest = tid >> 6;
    int s = rest % 2; rest /= 2;
    int t = rest % 7, a = rest / 7;
    int k = 4 * s + 2 * (l >> 4) + c;
    int n = 16 * t + (l & 15);
    W1f[tid] = (k < 6 && n < 100) ? W1[(a * 6 + k) * 100 + n] : 0.f;
  }
  if (tid < 3 * 25 * 64) {                         // W4f (single n-tile)
    int c = tid & 1, l = (tid >> 1) & 31, rest = tid >> 6;
    int s = rest % 25, a = rest / 25;
    int k = 4 * s + 2 * (l >> 4) + c;              // < 100 always
    int n = l & 15;
    W4f[tid] = (n < 6) ? W4[(a * 100 + k) * 6 + n] : 0.f;
  }
  if (tid < 3 * 112) {                             // b1p, b3p
    int n = tid % 112, a = tid / 112;
    b1p[tid] = (n < 100) ? b1[a * 100 + n] : 0.f;
    b3p[tid] = (n < 100) ? b3[a * 100 + n] : 0.f;
  }
  if (tid < 3 * 16) {                              // b4p
    int n = tid % 16, a = tid / 16;
    b4p[tid] = (n < 6) ? b4[a * 6 + n] : 0.f;
  }
}

// ---------------------------------------------------------------------------
// Kernel A: per-row argmax (first-max semantics), neighbor gather (att padded
// to stride 8 with zeros), and base copy out = batch_x. One wave32 per row.
// ---------------------------------------------------------------------------
__global__ __launch_bounds__(256) void sp_argmax_gather_copy(
    const float* __restrict__ x, float* __restrict__ out,
    float* __restrict__ attp, int* __restrict__ idxs, int Bn, int Xn)
{
  int wave = (blockIdx.x * blockDim.x + threadIdx.x) >> 5;
  int lane = threadIdx.x & 31;
  if (wave >= Bn) return;

  const float* row  = x   + (size_t)wave * Xn;
  float*       orow = out + (size_t)wave * Xn;

  float best = -INFINITY;
  int   bidx = 0x7fffffff;
  for (int i = lane; i < Xn; i += 32) {
    float v = row[i];
    orow[i] = v;                            // base copy (delta==0 cells)
    if (v > best) { best = v; bidx = i; }   // strict > keeps first max per lane
  }
  for (int off = 16; off >= 1; off >>= 1) { // tie -> smaller index
    float ov = __shfl_xor(best, off, 32);
    int   oi = __shfl_xor(bidx, off, 32);
    if (ov > best || (ov == best && oi < bidx)) { best = ov; bidx = oi; }
  }
  int ptr = bidx;

  if (lane < 8) {
    int idx;
    switch (lane) {
      case 0:  idx = 0;                 break;  // cell 0
      case 1:  idx = ptr;               break;  // pointer cell (unclamped)
      case 2:  idx = ptr - GRID_STRIDE; break;  // left
      case 3:  idx = ptr + GRID_STRIDE; break;  // right
      case 4:  idx = ptr - 1;           break;  // above
      case 5:  idx = ptr + 1;           break;  // below
      default: idx = 0;                 break;  // padding lanes 6,7
    }
    if (lane >= 2 && lane < 6) idx = min(max(idx, 1), Xn - 1);   // bind()
    attp[(size_t)wave * 8 + lane] = (lane < 6) ? row[idx] : 0.f;
    if (lane < 6) idxs[(size_t)wave * 6 + lane] = idx;
  }
}

// ---------------------------------------------------------------------------
// Kernel B: one wave32 per (16-sample tile, action). Runs 6->100->400->100->6
// MLP via V_WMMA_F32_16X16X4_F32; per K-step the loop body is exactly
// 1 ds_load_b64 (A) + 1 global_load_b64 (pre-swizzled B frag) + 1 v_wmma.
// Then scatters for samples whose action matches (ascending j => reference's
// "last index wins").
//
// fp32 WMMA layouts (ISA 7.12.2):
//   A 16x4 :  lane l -> M = l&15 ; VGPR v holds K = v + 2*(l>>4)
//   B  4x16:  lane l -> N = l&15 ; VGPR v holds K = v + 2*(l>>4)
//   C/D    :  VGPR r -> M = r + 8*(l>>4), N = l&15
// ---------------------------------------------------------------------------
__global__ __launch_bounds__(32) void sp_mlp_wmma_scatter(
    const float* __restrict__ x,
    const float* __restrict__ W1f, const float* __restrict__ b1p,
    const float* __restrict__ W2f, const float* __restrict__ b2,
    const float* __restrict__ W3f, const float* __restrict__ b3p,
    const float* __restrict__ W4f, const float* __restrict__ b4p,
    const int* __restrict__ act,
    const float* __restrict__ attp, const int* __restrict__ idxs,
    float* __restrict__ out, int Bn, int Xn)
{
  __shared__ float h1[16 * 112];   // layer1 activations; reused for layer3 out
  __shared__ float h2[16 * 400];   // layer2 activations
  __shared__ float pr[16 * 16];    // final pred tile (cols 0..5 valid)

  const int lane = threadIdx.x;
  const int a    = blockIdx.y;       // action 0..2
  const int tile = blockIdx.x;       // 16-sample tile
  const int m    = lane & 15;
  const int half = lane >> 4;
  const int row0 = tile * 16;

  // ---- layer 1: att[16x8] @ W1[8x112] + b1p, relu -> h1[16x112] -----------
  {
    const float* aRow = attp + (size_t)(row0 + m) * 8 + 2 * half;
    const float* wf   = W1f + (a * 7 * 2) * 64 + lane * 2;
#pragma unroll
    for (int t = 0; t < 7; ++t) {
      v8f acc = {};
#pragma unroll
      for (int s = 0; s < 2; ++s) {
        v2f av = *(const v2f*)(aRow + 4 * s);
        v2f bv = *(const v2f*)(wf + (t * 2 + s) * 64);
        acc = __builtin_amdgcn_wmma_f32_16x16x4_f32(false, av, false, bv,
                                                    (short)0, acc, false, false);
      }
      const float bias = b1p[a * 112 + t * 16 + m];
#pragma unroll
      for (int r = 0; r < 8; ++r) {
        float v = acc[r] + bias;
        h1[(r + 8 * half) * 112 + t * 16 + m] = v > 0.f ? v : 0.f;
      }
    }
  }

  // ---- layer 2: h1[16x100] @ W2[100x400] + b2, relu -> h2[16x400] ---------
  {
    const float* aRow = h1 + m * 112 + 2 * half;
    const float* wf0  = W2f + (a * 25 * 25) * 64 + lane * 2;
#pragma unroll 1
    for (int t = 0; t < 25; ++t) {
      const float* wf = wf0 + (t * 25) * 64;
      v8f acc = {};
#pragma unroll 5
      for (int s = 0; s < 25; ++s) {
        v2f av = *(const v2f*)(aRow + 4 * s);
        v2f bv = *(const v2f*)(wf + s * 64);
        acc = __builtin_amdgcn_wmma_f32_16x16x4_f32(false, av, false, bv,
                                                    (short)0, acc, false, false);
      }
      const float bias = b2[a * 400 + t * 16 + m];
#pragma unroll
      for (int r = 0; r < 8; ++r) {
        float v = acc[r] + bias;
        h2[(r + 8 * half) * 400 + t * 16 + m] = v > 0.f ? v : 0.f;
      }
    }
  }

  // ---- layer 3: h2[16x400] @ W3[400x112] + b3p, relu -> h1[16x112] --------
  {
    const float* aRow = h2 + m * 400 + 2 * half;
    const float* wf0  = W3f + (a * 7 * 100) * 64 + lane * 2;
#pragma unroll 1
    for (int t = 0; t < 7; ++t) {
      const float* wf = wf0 + (t * 100) * 64;
      v8f acc = {};
#pragma unroll 4
      for (int s = 0; s < 100; ++s) {
        v2f av = *(const v2f*)(aRow + 4 * s);
        v2f bv = *(const v2f*)(wf + s * 64);
        acc = __builtin_amdgcn_wmma_f32_16x16x4_f32(false, av, false, bv,
                                                    (short)0, acc, false, false);
      }
      const float bias = b3p[a * 112 + t * 16 + m];
#pragma unroll
      for (int r = 0; r < 8; ++r) {
        float v = acc[r] + bias;
        h1[(r + 8 * half) * 112 + t * 16 + m] = v > 0.f ? v : 0.f;
      }
    }
  }

  // ---- layer 4: h3[16x100] @ W4[100x16] + b4p -> pr[16x16] ----------------
  {
    const float* aRow = h1 + m * 112 + 2 * half;
    const float* wf   = W4f + (a * 25) * 64 + lane * 2;
    v8f acc = {};
#pragma unroll 5
    for (int s = 0; s < 25; ++s) {
      v2f av = *(const v2f*)(aRow + 4 * s);
      v2f bv = *(const v2f*)(wf + s * 64);
      acc = __builtin_amdgcn_wmma_f32_16x16x4_f32(false, av, false, bv,
                                                  (short)0, acc, false, false);
    }
    const float bias = b4p[a * 16 + m];
#pragma unroll
    for (int r = 0; r < 8; ++r)
      pr[(r + 8 * half) * 16 + m] = acc[r] + bias;
  }

  // ---- select by action + scatter (ascending j => last index wins) --------
  if (lane < 16) {
    const int rrow = row0 + lane;
    if (act[rrow] == a) {
#pragma unroll
      for (int j = 0; j < 6; ++j) {
        const int idx = idxs[(size_t)rrow * 6 + j];
        out[(size_t)rrow * Xn + idx] =
            x[(size_t)rrow * Xn + idx] + pr[lane * 16 + j];
      }
    }
  }
}

extern "C" void kernel_launch(void* const* d_in, const int* in_sizes, int n_in,
                              void* d_out, int out_size, void* d_ws, size_t ws_size,
                              hipStream_t stream) {
  const float* x  = (const float*)d_in[0];
  const float* W1 = (const float*)d_in[1];
  const float* b1 = (const float*)d_in[2];
  const float* W2 = (const float*)d_in[3];
  const float* b2 = (const float*)d_in[4];
  const float* W3 = (const float*)d_in[5];
  const float* b3 = (const float*)d_in[6];
  const float* W4 = (const float*)d_in[7];
  const float* b4 = (const float*)d_in[8];
  const int*  act = (const int*)d_in[9];

  const int Bn = in_sizes[9];            // 65536
  const int Xn = in_sizes[0] / Bn;       // 362
  float* out = (float*)d_out;

  // Workspace layout.
  char*  p    = (char*)d_ws;
  float* attp = (float*)p;               p += (size_t)Bn * 8 * sizeof(float);
  int*   idxs = (int*)p;                 p += (size_t)Bn * 6 * sizeof(int);
  float* W1f  = (float*)p;               p += 3 * 7 * 2 * 64 * sizeof(float);
  float* b1p  = (float*)p;               p += 3 * 112 * sizeof(float);
  float* W2f  = (float*)p;               p += 3 * 25 * 25 * 64 * sizeof(float);
  float* W3f  = (float*)p;               p += 3 * 7 * 100 * 64 * sizeof(float);
  float* b3p  = (float*)p;               p += 3 * 112 * sizeof(float);
  float* W4f  = (float*)p;               p += 3 * 25 * 64 * sizeof(float);
  float* b4p  = (float*)p;               p += 3 * 16 * sizeof(float);

  // Kernel P: repack weights (largest array: 3*7*100*64 = 134400 elems).
  sp_pack_weights<<<(3 * 7 * 100 * 64 + 255) / 256, 256, 0, stream>>>(
      W1, b1, W2, W3, b3, W4, b4, W1f, b1p, W2f, W3f, b3p, W4f, b4p);

  // Kernel A: 8 waves per 256-thread block, one wave per row.
  sp_argmax_gather_copy<<<(Bn + 7) / 8, 256, 0, stream>>>(
      x, out, attp, idxs, Bn, Xn);

  // Kernel B: one wave32 per (16-sample tile, action).
  dim3 gB((unsigned)(Bn / 16), 3);
  sp_mlp_wmma_scatter<<<gB, 32, 0, stream>>>(
      x, W1f, b1p, W2f, b2, W3f, b3p, W4f, b4p, act, attp, idxs, out, Bn, Xn);
}